// LTOCF_3118146257022
// MI455X (gfx1250) — compile-verified
//
#include <hip/hip_runtime.h>

#define LD 64

typedef __attribute__((ext_vector_type(2))) float v2f;
typedef __attribute__((ext_vector_type(8))) float v8f;

// ---- degree accumulation: deg[r] += 1 for both directions of each edge ----
__global__ __launch_bounds__(256) void deg_kernel(const int* __restrict__ eu,
                                                  const int* __restrict__ ei,
                                                  float* __restrict__ deg,
                                                  int E, int NU) {
    int e = blockIdx.x * blockDim.x + threadIdx.x;
    if (e >= E) return;
    atomicAdd(&deg[eu[e]], 1.0f);
    atomicAdd(&deg[ei[e] + NU], 1.0f);
}

// ---- dinv[n] = deg>0 ? rsqrt(deg) : 0 (in place) ----
__global__ __launch_bounds__(256) void dinv_kernel(float* __restrict__ deg, int N) {
    int n = blockIdx.x * blockDim.x + threadIdx.x;
    if (n >= N) return;
    float d = deg[n];
    deg[n] = (d > 0.0f) ? rsqrtf(d) : 0.0f;
}

// ---- x = acc = concat(user_emb, item_emb), float4 vectorized ----
__global__ __launch_bounds__(256) void init_kernel(const float* __restrict__ ue,
                                                   const float* __restrict__ ie,
                                                   float* __restrict__ x,
                                                   float* __restrict__ acc,
                                                   int nUser64, int total64) {
    int i = (blockIdx.x * blockDim.x + threadIdx.x) * 4;
    if (i >= total64) return;
    float4 v = (i < nUser64) ? *(const float4*)(ue + i)
                             : *(const float4*)(ie + (i - nUser64));
    *(float4*)(x + i)   = v;
    *(float4*)(acc + i) = v;
}

// ---- one wave per directed edge: y[r,:] += dinv[r]*dinv[c] * x[c,:] ----
// 32 lanes x float2 = one 64-float row; gathers/atomics hit the 192MB L2
// (entire 77MB working set is L2-resident on MI455X).
__global__ __launch_bounds__(256) void prop_kernel(const float* __restrict__ x,
                                                   float* __restrict__ y,
                                                   const int* __restrict__ eu,
                                                   const int* __restrict__ ei,
                                                   const float* __restrict__ dinv,
                                                   int E, int NU) {
    int gid  = blockIdx.x * blockDim.x + threadIdx.x;
    int wave = gid >> 5;
    int lane = threadIdx.x & 31;
    if (wave >= 2 * E) return;
    int r, c;
    if (wave < E) { r = eu[wave];          c = ei[wave] + NU; }
    else          { int e = wave - E; r = ei[e] + NU; c = eu[e]; }
    float w = dinv[r] * dinv[c];
    const float2 v = *(const float2*)(x + (size_t)c * LD + lane * 2);
    float* dst = y + (size_t)r * LD + lane * 2;
    atomicAdd(dst,     w * v.x);
    atomicAdd(dst + 1, w * v.y);
}

// ---- acc += y, float4 vectorized ----
__global__ __launch_bounds__(256) void accum_kernel(float* __restrict__ acc,
                                                    const float* __restrict__ y,
                                                    int total64) {
    int i = (blockIdx.x * blockDim.x + threadIdx.x) * 4;
    if (i >= total64) return;
    float4 a = *(const float4*)(acc + i);
    float4 b = *(const float4*)(y + i);
    a.x += b.x; a.y += b.y; a.z += b.z; a.w += b.w;
    *(float4*)(acc + i) = a;
}

// ---- batched row-dot via V_WMMA_F32_16X16X4_F32 ----
// One wave computes 16 outputs as the diagonal of a 16x16 U*I^T tile,
// accumulating K=64 in 16 steps of K=4 in the f32 WMMA accumulator.
// A layout (16x4 f32): lane = M (mod 16), half-wave -> K {0,1} vs {2,3}
// across the 2 VGPRs. B (4x16) mirrors with lane = N. D: VGPR j holds
// M=j (lanes 0-15, N=lane) and M=j+8 (lanes 16-31, N=lane-16).
__global__ __launch_bounds__(256) void dot_kernel(const float* __restrict__ acc,
                                                  const int* __restrict__ users,
                                                  const int* __restrict__ items,
                                                  float* __restrict__ out,
                                                  int NU, int B) {
    int gid  = blockIdx.x * blockDim.x + threadIdx.x;
    int wave = gid >> 5;
    int lane = threadIdx.x & 31;
    int base = wave * 16;
    if (base >= B) return;          // wave-uniform: EXEC stays all-ones for WMMA

    int mn = lane & 15;             // M for A, N for B
    int h  = lane >> 4;             // half-wave selects K sub-pair
    const float* urow = acc + (size_t)users[base + mn] * LD;
    const float* irow = acc + ((size_t)NU + (size_t)items[base + mn]) * LD;

    v8f c = {0.f, 0.f, 0.f, 0.f, 0.f, 0.f, 0.f, 0.f};
#pragma unroll
    for (int kk = 0; kk < 16; ++kk) {
        int k = kk * 4 + h * 2;
        float2 ua = *(const float2*)(urow + k);
        float2 ib = *(const float2*)(irow + k);
        v2f a; a[0] = ua.x; a[1] = ua.y;
        v2f b; b[0] = ib.x; b[1] = ib.y;
        c = __builtin_amdgcn_wmma_f32_16x16x4_f32(false, a, false, b,
                                                  (short)0, c, false, false);
    }

    // extract diagonal element for this lane (if it owns one)
    int j = (lane < 8) ? lane : (lane - 24);   // accumulator VGPR index
    float diag = 0.0f;
#pragma unroll
    for (int t = 0; t < 8; ++t) diag = (j == t) ? c[t] : diag;

    const float s = 1.0f / 25.0f;   // (mean over 5 stages) on both operands
    if (lane < 8)        out[base + lane]        = diag * s;
    else if (lane >= 24) out[base + (lane - 16)] = diag * s;
}

extern "C" void kernel_launch(void* const* d_in, const int* in_sizes, int n_in,
                              void* d_out, int out_size, void* d_ws, size_t ws_size,
                              hipStream_t stream) {
    const float* ue   = (const float*)d_in[0];
    const float* ie   = (const float*)d_in[1];
    const int* edge   = (const int*)d_in[2];
    const int* users  = (const int*)d_in[3];
    const int* items  = (const int*)d_in[4];

    const int NU = in_sizes[0] / LD;
    const int NI = in_sizes[1] / LD;
    const int E  = in_sizes[2] / 2;
    const int B  = in_sizes[3];
    const int N  = NU + NI;
    const int total64 = N * LD;

    const int* eu = edge;       // edge_index row 0
    const int* ei = edge + E;   // edge_index row 1

    // workspace layout (256B-aligned regions)
    float* ws   = (float*)d_ws;
    size_t dreg = ((size_t)N + 63) & ~(size_t)63;
    float* dinv = ws;
    float* xa   = ws + dreg;
    float* xb   = xa + (size_t)N * LD;
    float* acc  = xb + (size_t)N * LD;

    // 1) degrees -> dinv
    hipMemsetAsync(dinv, 0, (size_t)N * sizeof(float), stream);
    deg_kernel<<<(E + 255) / 256, 256, 0, stream>>>(eu, ei, dinv, E, NU);
    dinv_kernel<<<(N + 255) / 256, 256, 0, stream>>>(dinv, N);

    // 2) x = acc = e0
    init_kernel<<<((total64 / 4) + 255) / 256, 256, 0, stream>>>(
        ue, ie, xa, acc, NU * LD, total64);

    // 3) 4 propagation layers with running accumulation
    float* cur = xa;
    float* nxt = xb;
    for (int l = 0; l < 4; ++l) {
        hipMemsetAsync(nxt, 0, (size_t)total64 * sizeof(float), stream);
        long waves  = 2L * E;                 // one wave per directed edge
        int  blocks = (int)((waves + 7) / 8); // 8 waves (256 threads) per block
        prop_kernel<<<blocks, 256, 0, stream>>>(cur, nxt, eu, ei, dinv, E, NU);
        accum_kernel<<<((total64 / 4) + 255) / 256, 256, 0, stream>>>(acc, nxt, total64);
        float* t = cur; cur = nxt; nxt = t;
    }

    // 4) batched dots on the matrix core (16 outputs per wave)
    int dwaves = (B + 15) / 16;
    dot_kernel<<<(dwaves * 32 + 255) / 256, 256, 0, stream>>>(
        acc, users, items, (float*)d_out, NU, B);
}